// Decoder_89601607729224
// MI455X (gfx1250) — compile-verified
//
#include <hip/hip_runtime.h>
#include <hip/hip_bf16.h>

// Problem constants (from reference)
constexpr int S_ = 128;
constexpr int N_ = 256;
constexpr int H_ = 1024;
constexpr int E_ = 512;
constexpr int V_ = 32000;
constexpr int KDIM_ = 1024;
constexpr int RNNW = H_ + E_ + H_;   // 2560: [ctx | emb_x | h0]

typedef __attribute__((ext_vector_type(2))) float v2f;
typedef __attribute__((ext_vector_type(8))) float v8f;

// ---------------------------------------------------------------------------
// Block-wide sum reduction (8 waves of 32 lanes). All threads get the result.
// ---------------------------------------------------------------------------
__device__ inline float block_reduce_sum(float v, float* sm) {
#pragma unroll
    for (int o = 16; o > 0; o >>= 1) v += __shfl_down(v, o, 32);
    const int lane = threadIdx.x & 31;
    const int wid  = threadIdx.x >> 5;
    if (lane == 0) sm[wid] = v;
    __syncthreads();
    if (wid == 0) {
        float x = (lane < 8) ? sm[lane] : 0.0f;
#pragma unroll
        for (int o = 4; o > 0; o >>= 1) x += __shfl_down(x, o, 32);
        if (lane == 0) sm[8] = x;
    }
    __syncthreads();
    float r = sm[8];
    __syncthreads();
    return r;
}

// Batched 4-way block reduction: p[0..3] are summed over all 256 threads and
// broadcast back. One LDS round trip / 3 barriers for all four values.
__device__ inline void block_reduce_sum4(float p[4], float* sm) {
    const int lane = threadIdx.x & 31;
    const int wid  = threadIdx.x >> 5;
#pragma unroll
    for (int j = 0; j < 4; ++j) {
#pragma unroll
        for (int o = 16; o > 0; o >>= 1) p[j] += __shfl_down(p[j], o, 32);
    }
    if (lane == 0) {
#pragma unroll
        for (int j = 0; j < 4; ++j) sm[j * 8 + wid] = p[j];
    }
    __syncthreads();
    if (wid == 0) {
        const int j = lane >> 3;             // each 8-lane group sums one j
        const int i = lane & 7;
        float x = sm[j * 8 + i];
#pragma unroll
        for (int o = 4; o > 0; o >>= 1) x += __shfl_down(x, o, 8);
        if (i == 0) sm[32 + j] = x;
    }
    __syncthreads();
#pragma unroll
    for (int j = 0; j < 4; ++j) p[j] = sm[32 + j];
    __syncthreads();
}

// ---------------------------------------------------------------------------
// Kernel 1: fused energy(channel K-1) + online softmax over S + context.
// One block per batch element n; thread t owns H elements [4t, 4t+3].
// Reads encoder_states exactly once (134 MB). Processes 4 sequence rows per
// step (4-8 float4 loads in flight per lane, batched 4-way reduction) so the
// HBM pipe stays busy despite the per-step block reduction.
// Writes ctx into rnn_pack[n][0:H].
// ---------------------------------------------------------------------------
__global__ __launch_bounds__(256)
void attn_ctx_kernel(const float* __restrict__ enc,
                     const float* __restrict__ h0,
                     const float* __restrict__ W_energy,
                     const float* __restrict__ b_energy,
                     float* __restrict__ rnn_pack)
{
    __shared__ float sm[40];
    const int n  = blockIdx.x;
    const int t  = threadIdx.x;
    const int h4 = t << 2;

    const float* wrow = W_energy + (size_t)(KDIM_ - 1) * (2 * H_);
    const float4 weh  = *(const float4*)(wrow + h4);        // hidden half
    const float4 wee  = *(const float4*)(wrow + H_ + h4);   // encoder half
    const float4 h0v  = *(const float4*)(h0 + (size_t)n * H_ + h4);

    // dot(W_e[K-1][:H], h0[n]) + b  (shared by every s)
    float wh = block_reduce_sum(weh.x * h0v.x + weh.y * h0v.y +
                                weh.z * h0v.z + weh.w * h0v.w, sm)
             + b_energy[KDIM_ - 1];

    float m = -3.0e38f, d = 0.0f;
    float4 acc = {0.0f, 0.0f, 0.0f, 0.0f};

    const float* encn = enc + (size_t)n * H_ + h4;
    const size_t srow = (size_t)N_ * H_;

    float4 v[4];
#pragma unroll
    for (int j = 0; j < 4; ++j) v[j] = *(const float4*)(encn + j * srow);

    for (int s0 = 0; s0 < S_; s0 += 4) {
        float4 vn[4];
        if (s0 + 4 < S_) {                   // prefetch next group of 4 rows
#pragma unroll
            for (int j = 0; j < 4; ++j)
                vn[j] = *(const float4*)(encn + (size_t)(s0 + 4 + j) * srow);
        } else {
#pragma unroll
            for (int j = 0; j < 4; ++j) vn[j] = {0.0f, 0.0f, 0.0f, 0.0f};
        }
        float p[4];
#pragma unroll
        for (int j = 0; j < 4; ++j)
            p[j] = wee.x * v[j].x + wee.y * v[j].y +
                   wee.z * v[j].z + wee.w * v[j].w;
        block_reduce_sum4(p, sm);

#pragma unroll
        for (int j = 0; j < 4; ++j) {
            float e = fmaxf(p[j] + wh, 0.0f);   // relu(energy + bias)
            const float mn = fmaxf(m, e);
            const float sc = __expf(m - mn);    // first iter: exp(-inf)=0
            const float w  = __expf(e - mn);
            d = d * sc + w;
            acc.x = acc.x * sc + w * v[j].x;
            acc.y = acc.y * sc + w * v[j].y;
            acc.z = acc.z * sc + w * v[j].z;
            acc.w = acc.w * sc + w * v[j].w;
            m = mn;
        }
#pragma unroll
        for (int j = 0; j < 4; ++j) v[j] = vn[j];
    }
    const float inv = 1.0f / d;
    float* dst = rnn_pack + (size_t)n * RNNW + h4;
    float4 o = {acc.x * inv, acc.y * inv, acc.z * inv, acc.w * inv};
    *(float4*)dst = o;
}

// ---------------------------------------------------------------------------
// Kernel 2: finish packing rnn_in = [ctx | emb[input] | h0]  (cols H..RNNW)
// ---------------------------------------------------------------------------
__global__ __launch_bounds__(256)
void pack_rest_kernel(const int* __restrict__ input,
                      const float* __restrict__ emb,
                      const float* __restrict__ h0,
                      float* __restrict__ rnn_pack)
{
    const int n = blockIdx.x;
    const int t = threadIdx.x;
    const int row = input[n];
    float* dst = rnn_pack + (size_t)n * RNNW;
#pragma unroll
    for (int j = 0; j < 6; ++j) {
        const int c = t + (j << 8);          // 0..1535
        float v;
        if (c < E_) v = emb[(size_t)row * E_ + c];
        else        v = h0[(size_t)n * H_ + (c - E_)];
        dst[H_ + c] = v;
    }
}

// ---------------------------------------------------------------------------
// Kernel 3: generic f32 WMMA GEMM:  C[M,N'] (+)= A[M,K] @ B[N',K]^T + bias
//   grid = (N'/128, M/16), block = 256 (8 waves); one 16x16 tile per wave.
//   Staging uses CDNA5 async DMA (global_load_async_to_lds_b128, ASYNCcnt)
//   with double-buffered LDS tiles: chunk k+1 streams in while the WMMAs
//   consume chunk k. Row pitch 36 floats = 144 B: 16B-aligned rows for the
//   B128 LDS writes and bank-conflict-free for the 16x16x4 read pattern.
// ---------------------------------------------------------------------------
#define KC   32
#define LSTR 36

__global__ __launch_bounds__(256)
void gemm_wmma_f32(const float* __restrict__ A, int lda,
                   const float* __restrict__ B, int ldb,
                   float* __restrict__ C, int ldc,
                   int Ksz, const float* __restrict__ bias, int accumulate)
{
    __shared__ float ldsA[2][16 * LSTR];
    __shared__ float ldsB[2][128 * LSTR];

    const int t     = threadIdx.x;
    const int lane  = t & 31;
    const int wave  = t >> 5;
    const int mbase = blockIdx.y << 4;
    const int nbase = blockIdx.x << 7;

    const int am     = lane & 15;           // tile row (A) / tile col (B)
    const int khalf  = (lane >> 4) << 1;    // K offset 0 or 2 per half-wave
    const int rowoff = (lane >> 4) << 3;    // C/D: M offset 0 or 8
    const int col    = nbase + (wave << 4) + am;

    v8f acc;
    if (accumulate) {
#pragma unroll
        for (int r = 0; r < 8; ++r)
            acc[r] = C[(size_t)(mbase + r + rowoff) * ldc + col];
    } else {
#pragma unroll
        for (int r = 0; r < 8; ++r) acc[r] = 0.0f;
    }

    // Async DMA one K-chunk (16x32 A tile + 128x32 B tile) into LDS buf.
    auto stage_async = [&](int k0, int buf) {
        if (t < 128) {                       // A: waves 0-3, one B128 each
            const int r  = t >> 3;
            const int c4 = (t & 7) << 2;
            const float* g = A + (size_t)(mbase + r) * lda + k0 + c4;
            unsigned l = (unsigned)(size_t)&ldsA[buf][r * LSTR + c4];
            asm volatile("global_load_async_to_lds_b128 %0, %1, off"
                         :: "v"(l), "v"(g) : "memory");
        }
#pragma unroll
        for (int p = 0; p < 4; ++p) {        // B: 4 B128 per thread
            const int idx = p * 256 + t;
            const int r   = idx >> 3;
            const int c4  = (idx & 7) << 2;
            const float* g = B + (size_t)(nbase + r) * ldb + k0 + c4;
            unsigned l = (unsigned)(size_t)&ldsB[buf][r * LSTR + c4];
            asm volatile("global_load_async_to_lds_b128 %0, %1, off"
                         :: "v"(l), "v"(g) : "memory");
        }
    };

    const int nchunks = Ksz / KC;
    stage_async(0, 0);
    asm volatile("s_wait_asynccnt 0x0" ::: "memory");
    __syncthreads();

    for (int ch = 0; ch < nchunks; ++ch) {
        const int buf = ch & 1;
        if (ch + 1 < nchunks)                // overlap: fill other buffer
            stage_async((ch + 1) * KC, buf ^ 1);

        const float* bA = &ldsA[buf][am * LSTR + khalf];
        const float* bB = &ldsB[buf][((wave << 4) + am) * LSTR + khalf];
#pragma unroll
        for (int kk = 0; kk < KC; kk += 4) {
            v2f a, b;
            a.x = bA[kk]; a.y = bA[kk + 1];
            b.x = bB[kk]; b.y = bB[kk + 1];
            acc = __builtin_amdgcn_wmma_f32_16x16x4_f32(
                false, a, false, b, (short)0, acc, false, false);
        }
        asm volatile("s_wait_asynccnt 0x0" ::: "memory");
        __syncthreads();                     // next buffer ready; old one free
    }

    const float bv = bias ? bias[col] : 0.0f;
#pragma unroll
    for (int r = 0; r < 8; ++r)
        C[(size_t)(mbase + r + rowoff) * ldc + col] = acc[r] + bv;
}

// ---------------------------------------------------------------------------
// Kernel 4: LSTM pointwise (gate order i,f,g,o) -> h1, c1 (into d_out slots)
// ---------------------------------------------------------------------------
__global__ __launch_bounds__(256)
void lstm_pointwise(const float* __restrict__ gates,
                    const float* __restrict__ c0,
                    float* __restrict__ h1,
                    float* __restrict__ c1)
{
    const int idx = blockIdx.x * 256 + threadIdx.x;   // 0 .. N*H-1
    const int n = idx >> 10;
    const int h = idx & 1023;
    const float* g = gates + (size_t)n * (4 * H_);
    const float gi = g[h];
    const float gf = g[H_ + h];
    const float gg = g[2 * H_ + h];
    const float go = g[3 * H_ + h];
    const float si = 1.0f / (1.0f + __expf(-gi));
    const float sf = 1.0f / (1.0f + __expf(-gf));
    const float so = 1.0f / (1.0f + __expf(-go));
    const float c = sf * c0[idx] + si * tanhf(gg);
    c1[idx] = c;
    h1[idx] = so * tanhf(c);
}

// ---------------------------------------------------------------------------
extern "C" void kernel_launch(void* const* d_in, const int* in_sizes, int n_in,
                              void* d_out, int out_size, void* d_ws, size_t ws_size,
                              hipStream_t stream)
{
    (void)in_sizes; (void)n_in; (void)out_size; (void)ws_size;

    const int*   input    = (const int*)  d_in[0];
    const float* enc      = (const float*)d_in[1];
    const float* hidden   = (const float*)d_in[2];
    const float* cell     = (const float*)d_in[3];
    const float* emb      = (const float*)d_in[4];
    const float* W_energy = (const float*)d_in[5];
    const float* b_energy = (const float*)d_in[6];
    const float* W_ih     = (const float*)d_in[7];
    const float* b_ih     = (const float*)d_in[8];
    const float* W_hh     = (const float*)d_in[9];
    const float* b_hh     = (const float*)d_in[10];
    const float* W_fc     = (const float*)d_in[11];
    const float* b_fc     = (const float*)d_in[12];

    float* out  = (float*)d_out;
    float* pred = out;                                   // [N, V]
    float* h1   = out + (size_t)N_ * V_;                 // [N, H]
    float* c1   = h1 + (size_t)N_ * H_;                  // [N, H]

    float* ws       = (float*)d_ws;
    float* rnn_pack = ws;                                // [N, RNNW] = 2.6 MB
    float* gates    = ws + (size_t)N_ * RNNW;            // [N, 4H]   = 4.2 MB

    // 1) fused attention: energy ch K-1 -> online softmax -> ctx
    attn_ctx_kernel<<<N_, 256, 0, stream>>>(enc, hidden, W_energy, b_energy, rnn_pack);

    // 2) finish packing [ctx | emb_x | h0]
    pack_rest_kernel<<<N_, 256, 0, stream>>>(input, emb, hidden, rnn_pack);

    // 3) gates = rnn_in @ W_ih^T + b_ih   (K = H+E = 1536)
    gemm_wmma_f32<<<dim3(4 * H_ / 128, N_ / 16), 256, 0, stream>>>(
        rnn_pack, RNNW, W_ih, H_ + E_, gates, 4 * H_, H_ + E_, b_ih, 0);

    // 4) gates += h0 @ W_hh^T + b_hh      (K = H = 1024)
    gemm_wmma_f32<<<dim3(4 * H_ / 128, N_ / 16), 256, 0, stream>>>(
        rnn_pack + (H_ + E_), RNNW, W_hh, H_, gates, 4 * H_, H_, b_hh, 1);

    // 5) LSTM pointwise -> h1, c1 (directly into d_out)
    lstm_pointwise<<<(N_ * H_) / 256, 256, 0, stream>>>(gates, cell, h1, c1);

    // 6) predictions = h1 @ W_fc^T + b_fc (streams 131 MB of W_fc)
    gemm_wmma_f32<<<dim3(V_ / 128, N_ / 16), 256, 0, stream>>>(
        h1, H_, W_fc, H_, pred, V_, H_, b_fc, 0);
}